// MultiTypeAttentionDist2_87789131530408
// MI455X (gfx1250) — compile-verified
//
#include <hip/hip_runtime.h>
#include <hip/hip_bf16.h>
#include <math.h>

typedef float v2f __attribute__((ext_vector_type(2)));
typedef float v8f __attribute__((ext_vector_type(8)));

#define DIN  64
#define DRAW 64
#define DD   128
#define NEG_INF_C (-9000000000000000.0f)
#define NTHREADS 256

// One block per batch element b.
// Pass 1: e[n] = leaky(cb + raw_x[b,n,:].w) masked -> LDS, track max (WMMA f32 16x16x4)
// Pass 2: exp(e-m) in LDS, block-sum
// Pass 3: out = exp/sum * mask
__global__ __launch_bounds__(NTHREADS)
void MultiTypeAttention_kernel(const float* __restrict__ x,
                               const float* __restrict__ raw_x,
                               const float* __restrict__ a,
                               const float* __restrict__ adj,
                               const int*   __restrict__ node_index,
                               const int*   __restrict__ type_index,
                               float* __restrict__ out,
                               int N) {
    extern __shared__ float smem[];
    float* sh_e = smem;              // N floats: scores, then exp values
    float* sh_w = smem + N;          // DRAW floats: a[t, 64:128]
    float* sh_r = sh_w + DRAW;       // 257 floats: reductions + broadcast

    const int b    = blockIdx.x;
    const int tid  = threadIdx.x;
    const int t    = type_index[b];
    const int node = node_index[0];
    const float* at = a + (size_t)t * DD;

    // Load weight half into LDS; per-element products of the constant term.
    if (tid < DIN) {
        sh_w[tid] = at[DIN + tid];
        sh_r[tid] = x[((size_t)b * N + node) * (size_t)DIN + tid] * at[tid];
    }
    __syncthreads();
    if (tid == 0) {
        float c = 0.0f;
        for (int j = 0; j < DIN; ++j) c += sh_r[j];
        sh_r[256] = c;
    }
    __syncthreads();
    const float cb = sh_r[256];

    const int wave = tid >> 5;
    const int lane = tid & 31;
    const int half = lane >> 4;   // 0: K0/K1 half-wave, 1: K2/K3 half-wave
    const int col  = lane & 15;   // output column / row index n within tile

    // Hoist the loop-invariant A operand (w, replicated across M rows) into
    // registers: 16 v2f K-pairs selected by this lane's half-wave K split.
    v2f wreg[DRAW / 4];
    #pragma unroll
    for (int k = 0; k < DRAW / 4; ++k) {
        wreg[k] = *(const v2f*)(sh_w + 4 * k + half * 2);
    }
    __syncthreads();

    const float* rb = raw_x + (size_t)b * N * DRAW;
    float lmax = -INFINITY;

    // ---- Pass 1: scores via V_WMMA_F32_16X16X4_F32, 16 rows per wave-tile ----
    for (int n0 = wave * 16; n0 < N; n0 += 8 * 16) {
        if (n0 + 16 <= N) {   // wave-uniform: EXEC all-1s around WMMA
            // B operand: 16 raw_x rows, column-major. Lane covers row (n0+col),
            // K-pair {4k+2*half, 4k+2*half+1} per chunk.
            const float* rp = rb + (size_t)(n0 + col) * DRAW + half * 2;
            v8f acc = {};
            #pragma unroll
            for (int k = 0; k < DRAW / 4; ++k) {
                v2f bv = __builtin_nontemporal_load((const v2f*)(rp + 4 * k));
                acc = __builtin_amdgcn_wmma_f32_16x16x4_f32(
                        false, wreg[k], false, bv, (short)0, acc, false, false);
            }
            // A rows all equal w -> every D row holds the dot products; lane's
            // column is n0+col, so acc[0] == raw_x[b, n0+col, :] . w
            const int n = n0 + col;
            float e = acc[0] + cb;
            e = (e > 0.0f) ? e : 0.01f * e;                 // leaky relu
            const float mv = (adj[n] > 0.0f) ? e : NEG_INF_C;
            if (half == 0) sh_e[n] = mv;                    // lanes 16-31 are duplicates
            lmax = fmaxf(lmax, mv);
        } else {
            // Generic tail (unused for N % 16 == 0): scalar dot per lane.
            const int n = n0 + lane;
            if (n < N) {
                const float* rp = rb + (size_t)n * DRAW;
                float e = cb;
                for (int j = 0; j < DRAW; ++j) e += rp[j] * sh_w[j];
                e = (e > 0.0f) ? e : 0.01f * e;
                const float mv = (adj[n] > 0.0f) ? e : NEG_INF_C;
                sh_e[n] = mv;
                lmax = fmaxf(lmax, mv);
            }
        }
    }
    __syncthreads();

    // ---- Block max reduction ----
    sh_r[tid] = lmax;
    __syncthreads();
    for (int off = NTHREADS / 2; off > 0; off >>= 1) {
        if (tid < off) sh_r[tid] = fmaxf(sh_r[tid], sh_r[tid + off]);
        __syncthreads();
    }
    const float m = sh_r[0];
    __syncthreads();

    // ---- Pass 2: exponentiate in LDS, accumulate sum ----
    float lsum = 0.0f;
    for (int n = tid; n < N; n += NTHREADS) {
        const float ex = __expf(sh_e[n] - m);
        sh_e[n] = ex;
        lsum += ex;
    }
    __syncthreads();
    sh_r[tid] = lsum;
    __syncthreads();
    for (int off = NTHREADS / 2; off > 0; off >>= 1) {
        if (tid < off) sh_r[tid] += sh_r[tid + off];
        __syncthreads();
    }
    const float inv = 1.0f / sh_r[0];

    // ---- Pass 3: normalize, re-mask, store (streaming, non-temporal) ----
    float* ob = out + (size_t)b * N;
    for (int n = tid; n < N; n += NTHREADS) {
        __builtin_nontemporal_store(sh_e[n] * inv * adj[n], ob + n);
    }
}

extern "C" void kernel_launch(void* const* d_in, const int* in_sizes, int n_in,
                              void* d_out, int out_size, void* d_ws, size_t ws_size,
                              hipStream_t stream) {
    const float* x      = (const float*)d_in[0];   // (B, N, 64)
    const float* raw_x  = (const float*)d_in[1];   // (B, N, 64)
    const float* a      = (const float*)d_in[2];   // (T, 128, 1)
    const float* adj    = (const float*)d_in[3];   // (N,)
    const int*   node   = (const int*)d_in[4];     // scalar
    const int*   tindex = (const int*)d_in[5];     // (B,)
    float*       out    = (float*)d_out;           // (B, N)

    const int N = in_sizes[3];
    const int B = in_sizes[5];

    const size_t shbytes = ((size_t)N + DRAW + 257) * sizeof(float);

    MultiTypeAttention_kernel<<<dim3(B), dim3(NTHREADS), shbytes, stream>>>(
        x, raw_x, a, adj, node, tindex, out, N);
}